// GCN_62895501082738
// MI455X (gfx1250) — compile-verified
//
#include <hip/hip_runtime.h>

#define N_NODES 100000
#define N_EDGES 1600000
#define N_GRAPHS 64
#define FPAD 128
#define IN_FEATS 64

typedef __attribute__((ext_vector_type(2))) float v2f;
typedef __attribute__((ext_vector_type(8))) float v8f;

// ---------------------------------------------------------------- utilities

__global__ void fill0_v4(float4* __restrict__ p, int n4) {
    int i = blockIdx.x * blockDim.x + threadIdx.x;
    int stride = gridDim.x * blockDim.x;
    float4 z = make_float4(0.f, 0.f, 0.f, 0.f);
    for (; i < n4; i += stride) p[i] = z;
}

// x [N,64] -> h [N,128] zero padded
__global__ void pad_input(const float* __restrict__ x, float* __restrict__ h) {
    int i = blockIdx.x * blockDim.x + threadIdx.x;
    if (i >= N_NODES * FPAD) return;
    int n = i >> 7, c = i & (FPAD - 1);
    h[i] = (c < IN_FEATS) ? x[n * IN_FEATS + c] : 0.f;
}

// W [din,dout] -> Wp [128,128] zero padded (both self+neigh), b -> bp[128]
__global__ void pad_weights(const float* __restrict__ Ws, const float* __restrict__ Wn,
                            const float* __restrict__ b, int din, int dout,
                            float* __restrict__ WsP, float* __restrict__ WnP,
                            float* __restrict__ bP) {
    int i = blockIdx.x * blockDim.x + threadIdx.x;
    if (i >= FPAD * FPAD) return;
    int k = i >> 7, j = i & (FPAD - 1);
    bool in = (k < din) && (j < dout);
    WsP[i] = in ? Ws[k * dout + j] : 0.f;
    WnP[i] = in ? Wn[k * dout + j] : 0.f;
    if (i < FPAD) bP[i] = (i < dout) ? b[i] : 0.f;
}

__global__ void deg_kernel(const int* __restrict__ dst, float* __restrict__ deg) {
    int e = blockIdx.x * blockDim.x + threadIdx.x;
    if (e < N_EDGES) atomicAdd(&deg[dst[e]], 1.f);
}

__global__ void invdeg_kernel(const float* __restrict__ deg, float* __restrict__ invd) {
    int n = blockIdx.x * blockDim.x + threadIdx.x;
    if (n < N_NODES) {
        float d = deg[n];
        invd[n] = (d > 0.f) ? (1.f / d) : 0.f;
    }
}

// -------------------------------------------------------------- edge scatter
// one wave per edge; each lane moves 4 consecutive features (float4 load,
// 4x global_atomic_add_f32). kw = active feature width rounded up to 4.
__global__ void scatter_kernel(const float* __restrict__ h, const int* __restrict__ src,
                               const int* __restrict__ dst, float* __restrict__ agg,
                               int kw) {
    int e = blockIdx.x * (blockDim.x >> 5) + (threadIdx.x >> 5);
    int lane = threadIdx.x & 31;
    if (e >= N_EDGES) return;
    int s = src[e], d = dst[e];
    int c0 = lane << 2;
    if (c0 < kw) {
        float4 v = *reinterpret_cast<const float4*>(h + (size_t)s * FPAD + c0);
        float* ap = agg + (size_t)d * FPAD + c0;
        atomicAdd(ap + 0, v.x);
        atomicAdd(ap + 1, v.y);
        atomicAdd(ap + 2, v.z);
        atomicAdd(ap + 3, v.w);
    }
}

// --------------------------------------------------------------- WMMA GEMM
// out[n,j] = relu?( h[n,:]@Ws + (agg[n,:]*invd[n])@Wn + b[j] )
// Block = 256 threads = 8 waves; block handles one 16-row tile, each wave one
// 16-col tile. V_WMMA_F32_16X16X4_F32, both matmuls into one accumulator.
// A layout (ISA 7.12.2, 32-bit A 16x4): lanes 0-15 hold K={0,1}, lanes 16-31
// hold K={2,3}, M = lane&15.  B (4x16): lanes 0-15 N=lane K={0,1}, lanes
// 16-31 N=lane-16 K={2,3}.  C/D: VGPR r -> M = r + 8*(lane>=16), N = lane&15.
__global__ __launch_bounds__(256) void gemm_kernel(
        const float* __restrict__ hin, const float* __restrict__ agg,
        const float* __restrict__ invd,
        const float* __restrict__ WsP, const float* __restrict__ WnP,
        const float* __restrict__ bP,
        float* __restrict__ hout, int kIter, int applyRelu) {
    const int lane = threadIdx.x & 31;
    const int jt   = threadIdx.x >> 5;        // col tile 0..7
    const int m0   = blockIdx.x << 4;         // row tile base
    const int half = lane >> 4;               // 0 or 1
    const int ml   = lane & 15;
    const int row  = m0 + ml;                 // A-matrix row for this lane
    const int koff = half << 1;               // 0 or 2
    const int col  = (jt << 4) + ml;          // B/C column for this lane

    const float* hrow = hin + (size_t)row * FPAD;
    const float* arow = agg + (size_t)row * FPAD;
    const float  idg  = invd[row];

    v8f c = {};
    for (int ki = 0; ki < kIter; ++ki) {
        const int k0 = ki << 2;
        // self term
        v2f a;
        a.x = hrow[k0 + koff];
        a.y = hrow[k0 + koff + 1];
        v2f bm;
        bm.x = WsP[(k0 + koff) * FPAD + col];
        bm.y = WsP[(k0 + koff + 1) * FPAD + col];
        c = __builtin_amdgcn_wmma_f32_16x16x4_f32(false, a, false, bm,
                                                  (short)0, c, false, false);
        // mean-neighbor term
        v2f an;
        an.x = arow[k0 + koff] * idg;
        an.y = arow[k0 + koff + 1] * idg;
        v2f bn;
        bn.x = WnP[(k0 + koff) * FPAD + col];
        bn.y = WnP[(k0 + koff + 1) * FPAD + col];
        c = __builtin_amdgcn_wmma_f32_16x16x4_f32(false, an, false, bn,
                                                  (short)0, c, false, false);
    }
    const float bias = bP[col];
    float* obase = hout + (size_t)(m0 + (half << 3)) * FPAD + col;
#pragma unroll
    for (int r = 0; r < 8; ++r) {
        float v = c[r] + bias;
        if (applyRelu) v = fmaxf(v, 0.f);
        obase[(size_t)r * FPAD] = v;
    }
}

// --------------------------------------------------------- per-graph mean
__global__ void graph_reduce(const float* __restrict__ h, const int* __restrict__ gid,
                             float* __restrict__ gsum, float* __restrict__ gcnt) {
    __shared__ float ls[N_GRAPHS * 5];
    __shared__ float lc[N_GRAPHS];
    int t = threadIdx.x;
    for (int i = t; i < N_GRAPHS * 5; i += blockDim.x) ls[i] = 0.f;
    for (int i = t; i < N_GRAPHS; i += blockDim.x) lc[i] = 0.f;
    __syncthreads();
    int n = blockIdx.x * blockDim.x + t;
    if (n < N_NODES) {
        int g = gid[n];
        const float* hp = h + (size_t)n * FPAD;
#pragma unroll
        for (int cidx = 0; cidx < 5; ++cidx) atomicAdd(&ls[g * 5 + cidx], hp[cidx]);
        atomicAdd(&lc[g], 1.f);
    }
    __syncthreads();
    for (int i = t; i < N_GRAPHS * 5; i += blockDim.x)
        if (ls[i] != 0.f) atomicAdd(&gsum[i], ls[i]);
    for (int i = t; i < N_GRAPHS; i += blockDim.x)
        if (lc[i] != 0.f) atomicAdd(&gcnt[i], lc[i]);
}

__global__ void finalize_kernel(const float* __restrict__ gsum, const float* __restrict__ gcnt,
                                float* __restrict__ out) {
    int i = blockIdx.x * blockDim.x + threadIdx.x;
    if (i < N_GRAPHS * 5) out[i] = gsum[i] / fmaxf(gcnt[i / 5], 1.f);
}

// ------------------------------------------------------------------ launch

extern "C" void kernel_launch(void* const* d_in, const int* in_sizes, int n_in,
                              void* d_out, int out_size, void* d_ws, size_t ws_size,
                              hipStream_t stream) {
    const float* x   = (const float*)d_in[0];
    const int*   src = (const int*)d_in[1];
    const int*   dst = (const int*)d_in[2];
    const int*   gid = (const int*)d_in[3];
    const float* Ws[4] = {(const float*)d_in[4],  (const float*)d_in[7],
                          (const float*)d_in[10], (const float*)d_in[13]};
    const float* Wn[4] = {(const float*)d_in[5],  (const float*)d_in[8],
                          (const float*)d_in[11], (const float*)d_in[14]};
    const float* bb[4] = {(const float*)d_in[6],  (const float*)d_in[9],
                          (const float*)d_in[12], (const float*)d_in[15]};
    const int dins[4]  = {64, 128, 118, 103};
    const int douts[4] = {128, 118, 103, 5};

    float* hA   = (float*)d_ws;                         // [N,128]
    float* hB   = hA + (size_t)N_NODES * FPAD;          // [N,128]
    float* agg  = hB + (size_t)N_NODES * FPAD;          // [N,128]
    float* deg  = agg + (size_t)N_NODES * FPAD;         // [N]
    float* invd = deg + N_NODES;                        // [N]
    float* Wp   = invd + N_NODES;                       // 8 x 128*128
    float* bp   = Wp + (size_t)8 * FPAD * FPAD;         // 4 x 128
    float* gsum = bp + 4 * FPAD;                        // 64*5
    float* gcnt = gsum + N_GRAPHS * 5;                  // 64

    // prep: zero degree, pad input + weights
    fill0_v4<<<256, 256, 0, stream>>>((float4*)deg, N_NODES / 4);
    pad_input<<<(N_NODES * FPAD + 255) / 256, 256, 0, stream>>>(x, hA);
    for (int l = 0; l < 4; ++l)
        pad_weights<<<(FPAD * FPAD + 255) / 256, 256, 0, stream>>>(
            Ws[l], Wn[l], bb[l], dins[l], douts[l],
            Wp + (size_t)(2 * l) * FPAD * FPAD,
            Wp + (size_t)(2 * l + 1) * FPAD * FPAD,
            bp + l * FPAD);

    deg_kernel<<<(N_EDGES + 255) / 256, 256, 0, stream>>>(dst, deg);
    invdeg_kernel<<<(N_NODES + 255) / 256, 256, 0, stream>>>(deg, invd);

    float* hin = hA;
    float* hout = hB;
    for (int l = 0; l < 4; ++l) {
        fill0_v4<<<2048, 256, 0, stream>>>((float4*)agg, N_NODES * FPAD / 4);
        int kIter = (dins[l] + 3) / 4;
        scatter_kernel<<<N_EDGES / 8, 256, 0, stream>>>(hin, src, dst, agg, kIter * 4);
        gemm_kernel<<<N_NODES / 16, 256, 0, stream>>>(
            hin, agg, invd,
            Wp + (size_t)(2 * l) * FPAD * FPAD,
            Wp + (size_t)(2 * l + 1) * FPAD * FPAD,
            bp + l * FPAD,
            hout, kIter, (l < 3) ? 1 : 0);
        float* t = hin; hin = hout; hout = t;
    }

    // per-graph mean of final features (hin holds last output)
    fill0_v4<<<1, 96, 0, stream>>>((float4*)gsum, (N_GRAPHS * 5 + N_GRAPHS) / 4);
    graph_reduce<<<(N_NODES + 255) / 256, 256, 0, stream>>>(hin, gid, gsum, gcnt);
    finalize_kernel<<<2, 256, 0, stream>>>(gsum, gcnt, (float*)d_out);
}